// IonicPotential2D_23613730193757
// MI455X (gfx1250) — compile-verified
//
#include <hip/hip_runtime.h>
#include <math.h>

#define KEC 14.3996f
#define ALP 0.3f
#define NK 168
#define NKP 176
#define NIMG 25
#define NATOMS 256
#define NPAIRS 8192
#define CUT 6.0f
#define PI_F 3.14159265358979f
#define SQRT_PI_F 1.77245385090552f
#define TWO_PI_F 6.28318530717959f
#define TWOA_SQRTPI (2.0f*ALP/SQRT_PI_F)
#define FOURA_SQRTPI (4.0f*ALP/SQRT_PI_F)

typedef float v2f __attribute__((ext_vector_type(2)));
typedef float v8f __attribute__((ext_vector_type(8)));

// workspace layout (float offsets)
#define WS_YB    0      // 4
#define WS_SSUM  4
#define WS_REAL  5
#define WS_S2    6
#define WS_G     8      // 768 grad of total
#define WS_W     776    // 768 H*v
#define WS_RS    1544   // 768 shifted positions
#define WS_FF    2312   // 12
#define WS_FFN   2324   // 4
#define WS_V     2328   // 768
#define WS_KTAB  3096   // NKP*6 : hx,hy,hz,dh,1/dh,Ek
#define WS_THETA 4160   // 256*NKP

__device__ __forceinline__ float blockReduceSum(float v, float* sred) {
  int t = threadIdx.x;
  sred[t] = v; __syncthreads();
  for (int s = 128; s > 0; s >>= 1) {
    if (t < s) sred[t] += sred[t + s];
    __syncthreads();
  }
  float r = sred[0]; __syncthreads();
  return r;
}

__device__ __forceinline__ float areaOf(const float* cell) {
  float cx = cell[1]*cell[5] - cell[2]*cell[4];
  float cy = cell[2]*cell[3] - cell[0]*cell[5];
  float cz = cell[0]*cell[4] - cell[1]*cell[3];
  return sqrtf(cx*cx + cy*cy + cz*cz);
}

__global__ __launch_bounds__(256) void k_init(const float* R, const float* shift,
                                              const int* idx_m, const int* is_film,
                                              const float* recip, float* ws) {
  int t = threadIdx.x;
  for (int k = t; k < 8; k += 256) ws[k] = 0.f;
  for (int k = t; k < 768; k += 256) { ws[WS_G+k] = 0.f; ws[WS_W+k] = 0.f; }
  if (t < 16) ws[WS_FF + t] = 0.f;
  if (t < NATOMS) {
    float f = (is_film[t] > 0) ? 1.f : 0.f;
    int m = idx_m[t];
    ws[WS_RS+t*3+0] = R[t*3+0] + f*shift[m*3+0];
    ws[WS_RS+t*3+1] = R[t*3+1] + f*shift[m*3+1];
    ws[WS_RS+t*3+2] = R[t*3+2] + f*shift[m*3+2];
  }
  if (t < NKP) {
    float hx=0.f, hy=0.f, hz=0.f, dh=1.f, Ek=0.f;
    if (t < NK) {
      int lin = t + (t >= 84 ? 1 : 0);           // skip (0,0) at index 84
      float k0 = (float)(lin/13 - 6), k1 = (float)(lin%13 - 6);
      hx = TWO_PI_F*(k0*recip[0] + k1*recip[3]);
      hy = TWO_PI_F*(k0*recip[1] + k1*recip[4]);
      hz = TWO_PI_F*(k0*recip[2] + k1*recip[5]);
      dh = sqrtf(hx*hx + hy*hy + hz*hz);
      Ek = __expf(-dh*dh/(4.f*ALP*ALP));
    }
    float* kt = ws + WS_KTAB + t*6;
    kt[0]=hx; kt[1]=hy; kt[2]=hz; kt[3]=dh; kt[4]=1.f/dh; kt[5]=Ek;
  }
}

// Theta[j][k] = h_k . Rs_j via V_WMMA_F32_16X16X4_F32 tiles (M=k, N=atom, K=3 padded to 4)
__global__ __launch_bounds__(128) void k_theta(float* ws) {
  int wid  = blockIdx.x*4 + (threadIdx.x >> 5);   // 44*4 = 176 waves = 11*16 tiles
  int lane = threadIdx.x & 31;
  int ktile = wid % (NKP/16);
  int jtile = wid / (NKP/16);
  int m = lane & 15;
  bool lo = lane < 16;
  const float* kt = ws + WS_KTAB + (ktile*16 + m)*6;
  const float* rs = ws + WS_RS   + (jtile*16 + m)*3;
  v2f a, b;
  a.x = lo ? kt[0] : kt[2];   // K=0:hx | K=2:hz
  a.y = lo ? kt[1] : 0.f;     // K=1:hy | K=3:0
  b.x = lo ? rs[0] : rs[2];   // K=0:x  | K=2:z
  b.y = lo ? rs[1] : 0.f;     // K=1:y  | K=3:0
  v8f c = {0.f,0.f,0.f,0.f,0.f,0.f,0.f,0.f};
  c = __builtin_amdgcn_wmma_f32_16x16x4_f32(false, a, false, b, (short)0, c, false, false);
  int j = jtile*16 + m;
  int kbase = ktile*16 + (lo ? 0 : 8);
  float* th = ws + WS_THETA + j*NKP;
  #pragma unroll
  for (int r = 0; r < 8; ++r) th[kbase + r] = c[r];
}

__global__ __launch_bounds__(256) void k_born_force(const float* q, const float* offs,
                                                    const float* r0a, const float* na,
                                                    const int* ii, const int* jj,
                                                    const int* idx_m, float* ws) {
  int p = blockIdx.x*256 + threadIdx.x;
  if (p >= NPAIRS) return;
  int i = ii[p], j = jj[p];
  const float* Ri = ws + WS_RS + i*3;
  const float* Rj = ws + WS_RS + j*3;
  float rx = Rj[0]-Ri[0]+offs[p*3+0];
  float ry = Rj[1]-Ri[1]+offs[p*3+1];
  float rz = Rj[2]-Ri[2]+offs[p*3+2];
  float d = sqrtf(rx*rx + ry*ry + rz*rz);
  float n = na[p];
  float qq = fabsf(q[i]*q[j]);
  float B = qq*__powf(r0a[p], n-1.f)/n;
  if (d < CUT && d > 1e-6f) {
    float dpow = __powf(d, -n);
    float y = 0.5f*KEC*B*(dpow - __powf(CUT, -n));
    atomicAdd(ws + WS_YB + idx_m[i], y);
    float coef = -0.5f*KEC*B*n*dpow/(d*d);   // A'(d)/d ; gradU = coef*r
    atomicAdd(ws+WS_G+j*3+0,  coef*rx); atomicAdd(ws+WS_G+i*3+0, -coef*rx);
    atomicAdd(ws+WS_G+j*3+1,  coef*ry); atomicAdd(ws+WS_G+i*3+1, -coef*ry);
    atomicAdd(ws+WS_G+j*3+2,  coef*rz); atomicAdd(ws+WS_G+i*3+2, -coef*rz);
  }
}

__global__ __launch_bounds__(256) void k_coul_force(const float* q, const float* cell, float* ws) {
  __shared__ float sThI[NKP];
  __shared__ float sK[NKP*6];
  __shared__ float sAv[NIMG*3];
  __shared__ float sRed[256];
  int i = blockIdx.x, t = threadIdx.x, j = t;
  for (int k = t; k < NKP;   k += 256) sThI[k] = ws[WS_THETA + i*NKP + k];
  for (int k = t; k < NKP*6; k += 256) sK[k]   = ws[WS_KTAB + k];
  if (t < NIMG) {
    float n0 = (float)(t/5 - 2), n1 = (float)(t%5 - 2);
    sAv[t*3+0] = n0*cell[0] + n1*cell[3];
    sAv[t*3+1] = n0*cell[1] + n1*cell[4];
    sAv[t*3+2] = n0*cell[2] + n1*cell[5];
  }
  __syncthreads();
  const float* Ri = ws + WS_RS + i*3;
  const float* Rj = ws + WS_RS + j*3;
  float dx = Ri[0]-Rj[0], dy = Ri[1]-Rj[1], dz = Ri[2]-Rj[2];
  float qq = q[i]*q[j];
  float az = ALP*dz;
  float ez = __expf(-az*az);
  const float* thJ = ws + WS_THETA + j*NKP;
  float eS=0.f, fx=0.f, fy=0.f, fz=0.f;
  for (int k = 0; k < NK; ++k) {
    const float* kt = &sK[k*6];
    float p = sThI[k] - thJ[k];
    float sp, cp; __sincosf(p, &sp, &cp);
    float dh = kt[3], inv = kt[4];
    float hb = dh*(0.5f/ALP);
    float Xp = erfcxf(hb + az), Xm = erfcxf(hb - az);
    float E0 = kt[5]*ez;
    float g  = E0*(Xp + Xm);
    float gp = dh*E0*(Xp - Xm);
    float s  = qq*inv;
    eS += s*cp*g;
    fx -= s*sp*g*kt[0];
    fy -= s*sp*g*kt[1];
    fz += s*(cp*gp - sp*g*kt[2]);
  }
  float eR=0.f, gx=0.f, gy=0.f, gz=0.f;
  for (int im = 0; im < NIMG; ++im) {
    float rx = dx+sAv[im*3+0], ry = dy+sAv[im*3+1], rz = dz+sAv[im*3+2];
    float dsq = rx*rx + ry*ry + rz*rz;
    if (dsq > 1e-12f) {
      float d = sqrtf(dsq), invd = 1.f/d;
      float ad = ALP*d;
      float ec = erfcf(ad);
      float ex = __expf(-ad*ad);
      eR += qq*ec*invd;
      float phip = -TWOA_SQRTPI*ex*invd - ec*invd*invd;
      float cf = qq*phip*invd;
      gx += cf*rx; gy += cf*ry; gz += cf*rz;
    }
  }
  float erfaz = erff(az);
  float eS2 = qq*(dz*erfaz + ez*(1.f/(ALP*SQRT_PI_F)));
  float fS2z = qq*erfaz;
  float area = areaOf(cell);
  float cS = KEC*PI_F/(2.f*area), cS2f = 2.f*cS, cR = 0.5f*KEC;
  // weight 8 = 2 (ordered-pair symmetry) * 4 (N_MOL copies of yc in total)
  float Fx = 8.f*(cS*fx + cR*gx);
  float Fy = 8.f*(cS*fy + cR*gy);
  float Fz = 8.f*(cS*fz + cR*gz - cS2f*fS2z);
  float r0 = blockReduceSum(eS,  sRed);
  float r1 = blockReduceSum(eR,  sRed);
  float r2 = blockReduceSum(eS2, sRed);
  float r3 = blockReduceSum(Fx,  sRed);
  float r4 = blockReduceSum(Fy,  sRed);
  float r5 = blockReduceSum(Fz,  sRed);
  if (t == 0) {
    atomicAdd(ws+WS_SSUM, r0); atomicAdd(ws+WS_REAL, r1); atomicAdd(ws+WS_S2, r2);
    atomicAdd(ws+WS_G+i*3+0, r3); atomicAdd(ws+WS_G+i*3+1, r4); atomicAdd(ws+WS_G+i*3+2, r5);
  }
}

__global__ __launch_bounds__(256) void k_ff(const int* idx_m, const int* is_film, float* ws) {
  __shared__ float sff[12];
  int t = threadIdx.x;
  if (t < 12) sff[t] = 0.f;
  __syncthreads();
  float f = (is_film[t] > 0) ? 1.f : 0.f;
  int m = idx_m[t];
  atomicAdd(&sff[m*3+0], -ws[WS_G+t*3+0]*f);
  atomicAdd(&sff[m*3+1], -ws[WS_G+t*3+1]*f);
  atomicAdd(&sff[m*3+2], -ws[WS_G+t*3+2]*f);
  __syncthreads();
  ws[WS_V+t*3+0] = f*sff[m*3+0];
  ws[WS_V+t*3+1] = f*sff[m*3+1];
  ws[WS_V+t*3+2] = f*sff[m*3+2];
  if (t < 12) ws[WS_FF+t] = sff[t];
  if (t < 4)  ws[WS_FFN+t] = sff[t*3]*sff[t*3] + sff[t*3+1]*sff[t*3+1] + sff[t*3+2]*sff[t*3+2];
}

__global__ __launch_bounds__(256) void k_born_hvp(const float* q, const float* offs,
                                                  const float* r0a, const float* na,
                                                  const int* ii, const int* jj, float* ws) {
  int p = blockIdx.x*256 + threadIdx.x;
  if (p >= NPAIRS) return;
  int i = ii[p], j = jj[p];
  const float* Ri = ws + WS_RS + i*3;
  const float* Rj = ws + WS_RS + j*3;
  float rx = Rj[0]-Ri[0]+offs[p*3+0];
  float ry = Rj[1]-Ri[1]+offs[p*3+1];
  float rz = Rj[2]-Ri[2]+offs[p*3+2];
  float d = sqrtf(rx*rx + ry*ry + rz*rz);
  float n = na[p];
  float qq = fabsf(q[i]*q[j]);
  float B = qq*__powf(r0a[p], n-1.f)/n;
  if (d < CUT && d > 1e-6f) {
    float dpow = __powf(d, -n);
    float Ap_d = -0.5f*KEC*B*n*dpow/(d*d);          // A'/d
    float App  =  0.5f*KEC*B*n*(n+1.f)*dpow/(d*d);  // A''
    float dvx = ws[WS_V+j*3+0]-ws[WS_V+i*3+0];
    float dvy = ws[WS_V+j*3+1]-ws[WS_V+i*3+1];
    float dvz = ws[WS_V+j*3+2]-ws[WS_V+i*3+2];
    float rdv = rx*dvx + ry*dvy + rz*dvz;
    float c2 = (App - Ap_d)*rdv/(d*d);
    float wx = Ap_d*dvx + c2*rx;
    float wy = Ap_d*dvy + c2*ry;
    float wz = Ap_d*dvz + c2*rz;
    atomicAdd(ws+WS_W+j*3+0,  wx); atomicAdd(ws+WS_W+i*3+0, -wx);
    atomicAdd(ws+WS_W+j*3+1,  wy); atomicAdd(ws+WS_W+i*3+1, -wy);
    atomicAdd(ws+WS_W+j*3+2,  wz); atomicAdd(ws+WS_W+i*3+2, -wz);
  }
}

__global__ __launch_bounds__(256) void k_coul_hvp(const float* q, const float* cell, float* ws) {
  __shared__ float sThI[NKP];
  __shared__ float sK[NKP*6];
  __shared__ float sAv[NIMG*3];
  __shared__ float sRed[256];
  int i = blockIdx.x, t = threadIdx.x, j = t;
  for (int k = t; k < NKP;   k += 256) sThI[k] = ws[WS_THETA + i*NKP + k];
  for (int k = t; k < NKP*6; k += 256) sK[k]   = ws[WS_KTAB + k];
  if (t < NIMG) {
    float n0 = (float)(t/5 - 2), n1 = (float)(t%5 - 2);
    sAv[t*3+0] = n0*cell[0] + n1*cell[3];
    sAv[t*3+1] = n0*cell[1] + n1*cell[4];
    sAv[t*3+2] = n0*cell[2] + n1*cell[5];
  }
  __syncthreads();
  const float* Ri = ws + WS_RS + i*3;
  const float* Rj = ws + WS_RS + j*3;
  float dx = Ri[0]-Rj[0], dy = Ri[1]-Rj[1], dz = Ri[2]-Rj[2];
  float qq = q[i]*q[j];
  float az = ALP*dz;
  float ez = __expf(-az*az);
  float dvx = ws[WS_V+i*3+0]-ws[WS_V+j*3+0];
  float dvy = ws[WS_V+i*3+1]-ws[WS_V+j*3+1];
  float dvz = ws[WS_V+i*3+2]-ws[WS_V+j*3+2];
  const float* thJ = ws + WS_THETA + j*NKP;
  float wx=0.f, wy=0.f, wz=0.f;
  for (int k = 0; k < NK; ++k) {
    const float* kt = &sK[k*6];
    float p = sThI[k] - thJ[k];
    float sp, cp; __sincosf(p, &sp, &cp);
    float dh = kt[3], inv = kt[4];
    float hb = dh*(0.5f/ALP);
    float Xp = erfcxf(hb + az), Xm = erfcxf(hb - az);
    float E0 = kt[5]*ez;
    float g   = E0*(Xp + Xm);
    float gp  = dh*E0*(Xp - Xm);
    float gpp = dh*(dh*g - FOURA_SQRTPI*E0);
    float hdv = kt[0]*dvx + kt[1]*dvy + kt[2]*dvz;
    float s = qq*inv;
    float tA = cp*g*hdv + sp*gp*dvz;       // coefficient of -h
    wx -= s*tA*kt[0];
    wy -= s*tA*kt[1];
    wz += s*(-tA*kt[2] - sp*gp*hdv + cp*gpp*dvz);
  }
  float wRx=0.f, wRy=0.f, wRz=0.f;
  for (int im = 0; im < NIMG; ++im) {
    float rx = dx+sAv[im*3+0], ry = dy+sAv[im*3+1], rz = dz+sAv[im*3+2];
    float dsq = rx*rx + ry*ry + rz*rz;
    if (dsq > 1e-12f) {
      float d = sqrtf(dsq), invd = 1.f/d;
      float ad = ALP*d;
      float ec = erfcf(ad);
      float ex = __expf(-ad*ad);
      float phip  = -TWOA_SQRTPI*ex*invd - ec*invd*invd;
      float phipp =  TWOA_SQRTPI*ex*(2.f*ALP*ALP + 2.f*invd*invd) + 2.f*ec*invd*invd*invd;
      float rdv = rx*dvx + ry*dvy + rz*dvz;
      float c1 = qq*phip*invd;
      float c2 = qq*(phipp - phip*invd)*invd*invd*rdv;
      wRx += c1*dvx + c2*rx;
      wRy += c1*dvy + c2*ry;
      wRz += c1*dvz + c2*rz;
    }
  }
  float wS2z = qq*TWOA_SQRTPI*ez*dvz;      // psi''(z) = 2a/sqrt(pi) e^{-a^2 z^2}
  float area = areaOf(cell);
  float cS = KEC*PI_F/(2.f*area), cS2f = 2.f*cS, cR = 0.5f*KEC;
  float Wx = 8.f*(cS*wx + cR*wRx);
  float Wy = 8.f*(cS*wy + cR*wRy);
  float Wz = 8.f*(cS*wz + cR*wRz - cS2f*wS2z);
  float r3 = blockReduceSum(Wx, sRed);
  float r4 = blockReduceSum(Wy, sRed);
  float r5 = blockReduceSum(Wz, sRed);
  if (t == 0) {
    atomicAdd(ws+WS_W+i*3+0, r3);
    atomicAdd(ws+WS_W+i*3+1, r4);
    atomicAdd(ws+WS_W+i*3+2, r5);
  }
}

__global__ __launch_bounds__(256) void k_fin(const float* q, const float* cell,
                                             const int* idx_m, const int* is_film,
                                             float* ws, float* out) {
  __shared__ float sRed[256];
  __shared__ float sfng[12];
  int t = threadIdx.x;
  if (t < 12) sfng[t] = 0.f;
  __syncthreads();
  float f = (is_film[t] > 0) ? 1.f : 0.f;
  int m = idx_m[t];
  atomicAdd(&sfng[m*3+0], f*ws[WS_W+t*3+0]);
  atomicAdd(&sfng[m*3+1], f*ws[WS_W+t*3+1]);
  atomicAdd(&sfng[m*3+2], f*ws[WS_W+t*3+2]);
  float sq2 = blockReduceSum(q[t]*q[t], sRed);
  if (t == 0) {
    float area = areaOf(cell);
    float cS = KEC*PI_F/(2.f*area), cS2f = 2.f*cS, cR = 0.5f*KEC;
    float selfE = -(ALP/SQRT_PI_F)*sq2*KEC;
    float yc = cS*ws[WS_SSUM] - cS2f*ws[WS_S2] + cR*ws[WS_REAL] + selfE;
    out[4] = yc;
    for (int mm = 0; mm < 4; ++mm) {
      float yb = ws[WS_YB+mm];
      out[mm]   = yc + yb;   // y_energy
      out[5+mm] = yb;        // y_born
      out[9+mm] = ws[WS_FFN+mm];
    }
    for (int k = 0; k < 12; ++k) out[13+k] = -2.f*sfng[k];  // fng
  }
}

extern "C" void kernel_launch(void* const* d_in, const int* in_sizes, int n_in,
                              void* d_out, int out_size, void* d_ws, size_t ws_size,
                              hipStream_t stream) {
  (void)in_sizes; (void)n_in; (void)out_size; (void)ws_size;
  const float* R      = (const float*)d_in[0];
  const float* shift  = (const float*)d_in[1];
  const float* q      = (const float*)d_in[2];
  const float* offs   = (const float*)d_in[3];
  const float* cell   = (const float*)d_in[4];
  const float* recip  = (const float*)d_in[5];
  const float* r0a    = (const float*)d_in[6];
  const float* na     = (const float*)d_in[7];
  const int*   idx_i  = (const int*)d_in[8];
  const int*   idx_j  = (const int*)d_in[9];
  const int*   idx_m  = (const int*)d_in[10];
  const int*   isf    = (const int*)d_in[11];
  float* ws  = (float*)d_ws;
  float* out = (float*)d_out;

  k_init      <<<1,   256, 0, stream>>>(R, shift, idx_m, isf, recip, ws);
  k_theta     <<<44,  128, 0, stream>>>(ws);
  k_born_force<<<32,  256, 0, stream>>>(q, offs, r0a, na, idx_i, idx_j, idx_m, ws);
  k_coul_force<<<256, 256, 0, stream>>>(q, cell, ws);
  k_ff        <<<1,   256, 0, stream>>>(idx_m, isf, ws);
  k_born_hvp  <<<32,  256, 0, stream>>>(q, offs, r0a, na, idx_i, idx_j, ws);
  k_coul_hvp  <<<256, 256, 0, stream>>>(q, cell, ws);
  k_fin       <<<1,   256, 0, stream>>>(q, cell, idx_m, isf, ws, out);
}